// LightGCN_24790551233233
// MI455X (gfx1250) — compile-verified
//
#include <hip/hip_runtime.h>
#include <hip/hip_bf16.h>

#define N_USERS 100000
#define N_ITEMS 200000
#define N_NODES (N_USERS + N_ITEMS)
#define NNZ     3000000
#define DIM     64
#define M_PHI   1000
#define BATCH   8192

typedef __attribute__((ext_vector_type(2))) float v2f;
typedef __attribute__((ext_vector_type(8))) float v8f;

// ---------------------------------------------------------------- utilities

__global__ void k_zero(float* __restrict__ p, size_t n) {
    size_t i = (size_t)blockIdx.x * blockDim.x + threadIdx.x;
    if (i < n) p[i] = 0.0f;
}

// acc = all_emb + e   (all_emb = concat(user_emb, item_emb))
__global__ void k_acc_init(float* __restrict__ acc, const float* __restrict__ e,
                           const float* __restrict__ uemb, const float* __restrict__ iemb) {
    size_t i = (size_t)blockIdx.x * blockDim.x + threadIdx.x;
    if (i >= (size_t)N_NODES * DIM) return;
    size_t node = i / DIM;
    size_t d    = i - node * DIM;
    float base = (node < N_USERS) ? uemb[node * DIM + d]
                                  : iemb[(node - N_USERS) * DIM + d];
    acc[i] = base + e[i];
}

__global__ void k_acc_add(float* __restrict__ acc, const float* __restrict__ e, size_t n) {
    size_t i = (size_t)blockIdx.x * blockDim.x + threadIdx.x;
    if (i < n) acc[i] += e[i];
}

// ------------------------------------------------------------------ SpMM
// One wave (32 lanes) per nonzero, 2 floats per lane.
// nz is wave-uniform; readfirstlane pins the COO indices to SGPRs so the
// compiler uses the scalar (SMEM) path instead of 32 redundant lane loads.
// Scatter uses the native global_atomic_add_f32 (no-return, STOREcnt) via
// inline asm so it can never fall back to a CAS loop.
__global__ void k_spmm(const int* __restrict__ rows, const int* __restrict__ cols,
                       const float* __restrict__ vals,
                       const float* __restrict__ srcA, const float* __restrict__ srcB,
                       int splitN, float* __restrict__ dst) {
    size_t gid  = (size_t)blockIdx.x * blockDim.x + threadIdx.x;
    size_t nz   = gid >> 5;
    int    lane = (int)(gid & 31);
    if (nz >= NNZ) return;                 // wave-uniform guard (EXEC stays full)
    int nzu = __builtin_amdgcn_readfirstlane((int)nz);
    int   r = rows[nzu];
    int   c = cols[nzu];
    float v = vals[nzu];
    const float* src = (c < splitN) ? (srcA + (size_t)c * DIM)
                                    : (srcB + (size_t)(c - splitN) * DIM);
    float2 x = *(const float2*)(src + 2 * lane);   // coalesced 256B/wave, L2-resident
    float* d = dst + (size_t)r * DIM + 2 * lane;
    float f0 = v * x.x;
    float f1 = v * x.y;
    asm volatile(
        "global_atomic_add_f32 %0, %1, off\n\t"
        "global_atomic_add_f32 %0, %2, off offset:4"
        :: "v"(d), "v"(f0), "v"(f1)
        : "memory");
}

// ------------------------------------------------------- WMMA f32 GEMMs
// V_WMMA_F32_16X16X4_F32 fragment indexing (wave32):
//   A (16x4): lane<16 -> M=lane,    holds K=kk+0, kk+1
//             lane>=16 -> M=lane-16, holds K=kk+2, kk+3
//   B (4x16): same split on K, N = lane%16
//   C/D: VGPR r, lane<16 -> (M=r,   N=lane), lane>=16 -> (M=r+8, N=lane-16)

// P[64x64] = phi4[64 x 1000] @ ulm[1000 x 64].  One block, 16 waves = 16 tiles.
__global__ void k_gemm_P(const float* __restrict__ phi4, const float* __restrict__ ulm,
                         float* __restrict__ P) {
    int wave = threadIdx.x >> 5;
    int lane = threadIdx.x & 31;
    int mt = wave >> 2, nt = wave & 3;
    int lrow = lane & 15, hi = lane >> 4;
    int m = mt * 16 + lrow;
    int n = nt * 16 + lrow;
    v8f acc = {};
    for (int kk = 0; kk < M_PHI; kk += 4) {
        int k0 = kk + 2 * hi;
        v2f a = { phi4[(size_t)m * M_PHI + k0], phi4[(size_t)m * M_PHI + k0 + 1] };
        v2f b = { ulm[(size_t)k0 * DIM + n],   ulm[(size_t)(k0 + 1) * DIM + n] };
        acc = __builtin_amdgcn_wmma_f32_16x16x4_f32(false, a, false, b,
                                                    (short)0, acc, false, false);
    }
    for (int r = 0; r < 8; ++r) {
        int gm = mt * 16 + r + 8 * hi;
        P[(size_t)gm * DIM + nt * 16 + lrow] = acc[r];
    }
}

// user_lm_batch[8192 x 64] = 0.9*(phi2[users] @ ulm) + 0.1*(phi3[users] @ P)
// 2048 tiles (512 x 4), 8 waves/block -> 256 blocks.
__global__ void k_user_lm(const int* __restrict__ users,
                          const float* __restrict__ phi2, const float* __restrict__ phi3,
                          const float* __restrict__ ulm,  const float* __restrict__ P,
                          float* __restrict__ outlm) {
    int wave = threadIdx.x >> 5;
    int lane = threadIdx.x & 31;
    int tile = blockIdx.x * 8 + wave;
    int mt = tile >> 2, nt = tile & 3;
    int lrow = lane & 15, hi = lane >> 4;
    int u = users[mt * 16 + lrow];
    const float* p2 = phi2 + (size_t)u * M_PHI;
    const float* p3 = phi3 + (size_t)u * DIM;
    int n = nt * 16 + lrow;

    v8f acc1 = {};                       // phi2 @ ulm, K = 1000
    for (int kk = 0; kk < M_PHI; kk += 4) {
        int k0 = kk + 2 * hi;
        v2f a = { p2[k0], p2[k0 + 1] };
        v2f b = { ulm[(size_t)k0 * DIM + n], ulm[(size_t)(k0 + 1) * DIM + n] };
        acc1 = __builtin_amdgcn_wmma_f32_16x16x4_f32(false, a, false, b,
                                                     (short)0, acc1, false, false);
    }
    v8f acc2 = {};                       // phi3 @ P, K = 64
    for (int kk = 0; kk < DIM; kk += 4) {
        int k0 = kk + 2 * hi;
        v2f a = { p3[k0], p3[k0 + 1] };
        v2f b = { P[k0 * DIM + n], P[(k0 + 1) * DIM + n] };
        acc2 = __builtin_amdgcn_wmma_f32_16x16x4_f32(false, a, false, b,
                                                     (short)0, acc2, false, false);
    }
    for (int r = 0; r < 8; ++r) {
        int gm = mt * 16 + r + 8 * hi;
        outlm[(size_t)gm * DIM + nt * 16 + lrow] = 0.9f * acc1[r] + 0.1f * acc2[r];
    }
}

// --------------------------------------------------------------- final dots
// One wave per batch element; light_out = acc/4 and the 0.5 factor fold to 1/32.
__global__ void k_final(const int* __restrict__ users, const int* __restrict__ items,
                        const float* __restrict__ acc,
                        const float* __restrict__ ulm_batch,
                        const float* __restrict__ item_lm,
                        float* __restrict__ out) {
    size_t gid = (size_t)blockIdx.x * blockDim.x + threadIdx.x;
    int b    = (int)(gid >> 5);
    int lane = (int)(gid & 31);
    if (b >= BATCH) return;
    int u  = users[b];
    int it = items[b];
    float2 x = ((const float2*)(acc + (size_t)u * DIM))[lane];
    float2 y = ((const float2*)(acc + (size_t)(N_USERS + it) * DIM))[lane];
    float sid = x.x * y.x + x.y * y.y;
    float2 p = ((const float2*)(ulm_batch + (size_t)b * DIM))[lane];
    float2 q = ((const float2*)(item_lm + (size_t)it * DIM))[lane];
    float slm = p.x * q.x + p.y * q.y;
    for (int off = 16; off > 0; off >>= 1) {
        sid += __shfl_down(sid, off, 32);
        slm += __shfl_down(slm, off, 32);
    }
    if (lane == 0) out[b] = 0.03125f * sid + slm;
}

// ------------------------------------------------------------------ launch
extern "C" void kernel_launch(void* const* d_in, const int* in_sizes, int n_in,
                              void* d_out, int out_size, void* d_ws, size_t ws_size,
                              hipStream_t stream) {
    const int*   users    = (const int*)  d_in[0];
    const int*   items    = (const int*)  d_in[1];
    const float* user_emb = (const float*)d_in[2];
    const float* item_emb = (const float*)d_in[3];
    const int*   rows     = (const int*)  d_in[4];
    const int*   cols     = (const int*)  d_in[5];
    const float* vals     = (const float*)d_in[6];
    const float* phi3     = (const float*)d_in[7];
    const float* phi4     = (const float*)d_in[8];
    const float* phi2     = (const float*)d_in[9];
    const float* ulm      = (const float*)d_in[10];
    const float* item_lm  = (const float*)d_in[11];
    float* out = (float*)d_out;

    const size_t nodef = (size_t)N_NODES * DIM;
    float* e0        = (float*)d_ws;
    float* e1        = e0 + nodef;
    float* acc       = e1 + nodef;
    float* P         = acc + nodef;
    float* ulm_batch = P + (size_t)DIM * DIM;

    const int TPB = 256;
    const unsigned elem_blocks = (unsigned)((nodef + TPB - 1) / TPB);
    const unsigned spmm_blocks = (unsigned)(((size_t)NNZ * 32 + TPB - 1) / TPB);

    // layer 1: e0 = spmm(all_emb); acc = all_emb + e0
    k_zero<<<elem_blocks, TPB, 0, stream>>>(e0, nodef);
    k_spmm<<<spmm_blocks, TPB, 0, stream>>>(rows, cols, vals, user_emb, item_emb, N_USERS, e0);
    k_acc_init<<<elem_blocks, TPB, 0, stream>>>(acc, e0, user_emb, item_emb);
    // layer 2: e1 = spmm(e0); acc += e1
    k_zero<<<elem_blocks, TPB, 0, stream>>>(e1, nodef);
    k_spmm<<<spmm_blocks, TPB, 0, stream>>>(rows, cols, vals, e0, e0, N_NODES, e1);
    k_acc_add<<<elem_blocks, TPB, 0, stream>>>(acc, e1, nodef);
    // layer 3: e0 = spmm(e1); acc += e0
    k_zero<<<elem_blocks, TPB, 0, stream>>>(e0, nodef);
    k_spmm<<<spmm_blocks, TPB, 0, stream>>>(rows, cols, vals, e1, e1, N_NODES, e0);
    k_acc_add<<<elem_blocks, TPB, 0, stream>>>(acc, e0, nodef);

    // LM path: P = phi4 @ ulm;  user_lm for the batch only (WMMA)
    k_gemm_P<<<1, 512, 0, stream>>>(phi4, ulm, P);
    k_user_lm<<<(BATCH / 16) * (DIM / 16) / 8, 256, 0, stream>>>(users, phi2, phi3, ulm, P, ulm_batch);

    // final: out = 0.5*gamma_id + gamma_lm
    k_final<<<(BATCH * 32) / TPB, TPB, 0, stream>>>(users, items, acc, ulm_batch, item_lm, out);
}